// CenterLoss_56401510531693
// MI455X (gfx1250) — compile-verified
//
#include <hip/hip_runtime.h>

typedef float v2f __attribute__((ext_vector_type(2)));
typedef float v4f __attribute__((ext_vector_type(4)));
typedef float v8f __attribute__((ext_vector_type(8)));

#define NUM_CLASSES 10000
#define FEAT_DIM    512
#define BATCH       4096
#define CLAMP_MIN   1e-12f
#define CLAMP_MAX   1e12f

#define TILE            16   // samples per wave (one WMMA tile edge)
#define WAVES_PER_BLOCK 8
#define BLOCK_THREADS   (WAVES_PER_BLOCK * 32)
#define SAMPLES_PER_BLOCK (WAVES_PER_BLOCK * TILE)   // 128
#define NUM_TILES       (BATCH / TILE)               // 256

// Stage 1: each wave computes 16 per-sample squared distances via
// V_WMMA_F32_16X16X4_F32 (diagonal of A * B^T over gathered centers),
// writing one partial sum per 16-sample tile.
//
// K-window permutation trick: a single b128 load per lane supplies TWO
// 16x16x4 WMMAs. Lane group g (lanes 16*g..16*g+15) loads mem columns
// [k0+4g, k0+4g+3]. WMMA step 0 consumes {a.x,a.y} (virtual K = mem cols
// {k0,k0+1,k0+4,k0+5}), step 1 consumes {a.z,a.w}. A and B use the same
// permutation, so the accumulated dot product is exact.
__global__ __launch_bounds__(BLOCK_THREADS)
void center_loss_tile_kernel(const float* __restrict__ x,
                             const int*   __restrict__ labels,
                             const float* __restrict__ centers,
                             float*       __restrict__ partial)
{
    __shared__ float diag_lds[WAVES_PER_BLOCK][32 * 8];

    const int lane = threadIdx.x & 31;
    const int wave = threadIdx.x >> 5;
    const int tile = blockIdx.x * WAVES_PER_BLOCK + wave;

    const int n = lane & 15;   // sample/column index within the 16-wide tile
    const int g = lane >> 4;   // K sub-group selector

    const int row = tile * TILE + n;
    const int lbl = labels[row];

    const float* __restrict__ xrow = x       + (size_t)row * FEAT_DIM;
    const float* __restrict__ crow = centers + (size_t)lbl * FEAT_DIM;

    v8f acc  = {};          // 16x16 f32 accumulator tile (8 VGPRs)
    v4f xacc = {};          // packed partial of ||x_row||^2
    v4f cacc = {};          // packed partial of ||c_lbl||^2

    #pragma unroll 2
    for (int k0 = 0; k0 < FEAT_DIM; k0 += 8) {
        v4f a = *(const v4f*)(xrow + k0 + 4 * g);
        v4f b = *(const v4f*)(crow + k0 + 4 * g);

        xacc += a * a;       // -> v_pk_fma_f32 x2, no reg shuffling
        cacc += b * b;

        // (neg_a, A, neg_b, B, c_mod, C, reuse_a, reuse_b)
        acc = __builtin_amdgcn_wmma_f32_16x16x4_f32(
                  false, a.xy, false, b.xy, (short)0, acc, false, false);
        acc = __builtin_amdgcn_wmma_f32_16x16x4_f32(
                  false, a.zw, false, b.zw, (short)0, acc, false, false);
    }

    // Horizontal fold of packed norm partials, then fold the two lane groups.
    float xsq = (xacc.x + xacc.y) + (xacc.z + xacc.w);
    float csq = (cacc.x + cacc.y) + (cacc.z + cacc.w);
    const float xs = xsq + __shfl_xor(xsq, 16);
    const float cs = csq + __shfl_xor(csq, 16);

    // Spill accumulator tile to LDS to pull out the diagonal.
    #pragma unroll
    for (int r = 0; r < 8; ++r)
        diag_lds[wave][lane * 8 + r] = acc[r];
    __syncthreads();

    // D layout: VGPR r, lanes 0-15 -> (M=r, N=lane); lanes 16-31 -> (M=r+8, N=lane-16)
    // diag m<8  : VGPR m,   lane m
    // diag m>=8 : VGPR m-8, lane m+16
    float d = 0.0f;
    if (lane < TILE) {
        const int m = lane;
        const float dot = (m < 8) ? diag_lds[wave][m * 8 + m]
                                  : diag_lds[wave][(m + 16) * 8 + (m - 8)];
        d = xs + cs - 2.0f * dot;           // distmat[b, label_b]
        d = fminf(fmaxf(d, CLAMP_MIN), CLAMP_MAX);
    }

    // Deterministic intra-wave tree sum (lanes >= 16 contribute 0).
    #pragma unroll
    for (int off = 16; off >= 1; off >>= 1)
        d += __shfl_xor(d, off);

    if (lane == 0)
        partial[tile] = d;
}

// Stage 2: reduce the 256 tile partials, add the masked-zero clamp floor,
// divide by batch size.
__global__ __launch_bounds__(256)
void center_loss_reduce_kernel(const float* __restrict__ partial,
                               float*       __restrict__ out)
{
    __shared__ float s[NUM_TILES];
    const int t = threadIdx.x;
    s[t] = partial[t];
    __syncthreads();
    #pragma unroll
    for (int stride = NUM_TILES / 2; stride >= 1; stride >>= 1) {
        if (t < stride) s[t] += s[t + stride];
        __syncthreads();
    }
    if (t == 0) {
        // Every masked-out entry clamps up to 1e-12: (C-1) of them per row.
        out[0] = s[0] / (float)BATCH + (float)(NUM_CLASSES - 1) * CLAMP_MIN;
    }
}

extern "C" void kernel_launch(void* const* d_in, const int* in_sizes, int n_in,
                              void* d_out, int out_size, void* d_ws, size_t ws_size,
                              hipStream_t stream)
{
    const float* x       = (const float*)d_in[0];
    const int*   labels  = (const int*)  d_in[1];
    const float* centers = (const float*)d_in[2];
    float*       out     = (float*)d_out;
    float*       partial = (float*)d_ws;      // NUM_TILES floats of scratch

    const int grid1 = BATCH / SAMPLES_PER_BLOCK;   // 32 blocks of 8 waves
    center_loss_tile_kernel<<<grid1, BLOCK_THREADS, 0, stream>>>(
        x, labels, centers, partial);

    center_loss_reduce_kernel<<<1, NUM_TILES, 0, stream>>>(partial, out);
}